// GrfSim_83545703841917
// MI455X (gfx1250) — compile-verified
//
#include <hip/hip_runtime.h>
#include <math.h>

// ---------------------------------------------------------------------------
// Graph-matching MPM (N=80, 50 iters) as a single persistent-workgroup kernel.
//   M[j,a]  = max_b x[j,b]*tgt[a,b]        (tropical product, VALU)
//   neigh   = src @ M                      (dense f32 GEMM -> v_wmma_f32_16x16x4_f32)
//   x       = (x*d + neigh) / ||.||_F      (elementwise + deterministic reduction)
// Whole state lives in LDS (~133 KB of the WGP's 320 KB); no HBM traffic
// between iterations. 25 wave32 waves, one 16x16 tile each.
// ---------------------------------------------------------------------------

#define NN       80
#define MPM_IT   50
#define STRIDE   84   // row padding: 84 mod 64 = 20 -> 16 distinct LDS banks across a half-wave
#define NWAVES   25
#define NTHREADS 800

typedef float v2f __attribute__((ext_vector_type(2)));
typedef float v8f __attribute__((ext_vector_type(8)));

__global__ __launch_bounds__(NTHREADS)
void grf_mpm_kernel(const float* __restrict__ recon,
                    const float* __restrict__ adj,
                    float* __restrict__ out)
{
    extern __shared__ float smem[];
    float* s_src = smem;                    // [NN][STRIDE]
    float* s_tgt = s_src + NN * STRIDE;     // [NN][STRIDE]
    float* s_d   = s_tgt + NN * STRIDE;     // [NN][STRIDE]
    float* s_x   = s_d   + NN * STRIDE;     // [NN][STRIDE]
    float* s_M   = s_x   + NN * STRIDE;     // [NN][STRIDE]
    float* s_af  = s_M   + NN * STRIDE;     // adj row sums      [NN]
    float* s_of  = s_af  + NN;              // recon row sums    [NN]
    float* s_ad  = s_of  + NN;              // adj diagonal      [NN]
    float* s_rd  = s_ad  + NN;              // recon diagonal    [NN]
    float* s_red = s_rd  + NN;              // wave partials     [32]
    float* s_sc  = s_red + 32;              // scalars           [4]

    const int tid  = threadIdx.x;
    const int lane = tid & 31;
    const int wave = tid >> 5;

    // ---- setup: row sums + diagonals ------------------------------------
    if (tid < NN) {
        float sa = 0.f, sr = 0.f;
        for (int c = 0; c < NN; ++c) {
            sa += adj[tid * NN + c];
            sr += recon[tid * NN + c];
        }
        s_af[tid] = sa;                       // adj_features[i]
        s_of[tid] = sr;                       // out_features[a]
        s_ad[tid] = adj[tid * NN + tid];      // adj[i,i]
        s_rd[tid] = recon[tid * NN + tid];    // recon[a,a]
    }
    __syncthreads();

    // ---- build src / tgt / d / x0 in LDS; accumulate BCE partial --------
    float bce = 0.f;
    for (int idx = tid; idx < NN * NN; idx += NTHREADS) {
        int i = idx / NN;
        int j = idx - i * NN;
        float ta = adj[idx];
        float tr = recon[idx];
        float offd = (i == j) ? 0.f : 1.f;
        s_src[i * STRIDE + j] = ta * s_ad[i] * s_ad[j] * offd;
        s_tgt[i * STRIDE + j] = tr * s_rd[i] * s_rd[j] * offd;
        s_d  [i * STRIDE + j] = s_ad[i] * s_rd[j] / (fabsf(s_af[i] - s_of[j]) + 1.f);
        s_x  [i * STRIDE + j] = 1.f / (float)NN;
        if (j >= i) {  // upper triangle incl. diagonal
            bce -= ta * logf(tr) + (1.f - ta) * logf(1.f - tr);
        }
    }
    // deterministic BCE reduction: wave shuffle -> LDS -> wave 0
    #pragma unroll
    for (int off = 16; off; off >>= 1) bce += __shfl_xor(bce, off, 32);
    if (lane == 0) s_red[wave] = bce;
    __syncthreads();
    if (wave == 0) {
        float t = (lane < NWAVES) ? s_red[lane] : 0.f;
        #pragma unroll
        for (int off = 16; off; off >>= 1) t += __shfl_xor(t, off, 32);
        if (lane == 0) out[0] = t / (float)(NN * (NN + 1) / 2);
    }
    __syncthreads();

    // ---- tile assignment: 25 waves x one 16x16 tile ---------------------
    const int wr = wave / 5;
    const int wc = wave - 5 * wr;
    const int r0 = wr * 16;
    const int c0 = wc * 16;
    const int hh = lane >> 4;        // half-wave: 0 or 1
    const int m  = lane & 15;        // 0..15
    const int kk = 2 * hh;           // per-ISA f32 WMMA A/B: K = 2*half + vgpr

    for (int it = 0; it < MPM_IT; ++it) {
        // Phase A: tropical product  M[j,a] = max_b x[j,b]*tgt[a,b]
        // (lane layout mirrors the WMMA C fragment: row = v + 8*hh, col = m)
        #pragma unroll
        for (int v = 0; v < 8; ++v) {
            int j = r0 + v + 8 * hh;
            int a = c0 + m;
            const float* xr = s_x   + j * STRIDE;   // broadcast across half-wave
            const float* tg = s_tgt + a * STRIDE;   // stride-84 -> conflict-free
            float best = xr[0] * tg[0];
            #pragma unroll 4
            for (int b = 1; b < NN; ++b) best = fmaxf(best, xr[b] * tg[b]);
            s_M[j * STRIDE + a] = best;
        }
        __syncthreads();

        // Phase B: neigh(16x16) = src(16x80) @ M(80x16), K-chunks of 4
        v8f acc = {0.f, 0.f, 0.f, 0.f, 0.f, 0.f, 0.f, 0.f};
        for (int kc = 0; kc < NN; kc += 4) {
            v2f av, bv;
            // A fragment (src): lane (hh,m), VGPR v -> A[r0+m][kc + 2*hh + v]
            av[0] = s_src[(r0 + m) * STRIDE + kc + kk + 0];
            av[1] = s_src[(r0 + m) * STRIDE + kc + kk + 1];
            // B fragment (M): lane (hh,m), VGPR v -> B[kc + 2*hh + v][c0+m]
            bv[0] = s_M[(kc + kk + 0) * STRIDE + c0 + m];
            bv[1] = s_M[(kc + kk + 1) * STRIDE + c0 + m];
            acc = __builtin_amdgcn_wmma_f32_16x16x4_f32(
                false, av, false, bv, (short)0, acc, false, false);
        }

        // Phase C: x_new = x*d + neigh ; Frobenius norm ; rescale
        float xn[8];
        float ssq = 0.f;
        #pragma unroll
        for (int v = 0; v < 8; ++v) {
            int i = r0 + v + 8 * hh;
            int a = c0 + m;
            float val = s_x[i * STRIDE + a] * s_d[i * STRIDE + a] + acc[v];
            xn[v] = val;
            ssq += val * val;
        }
        #pragma unroll
        for (int off = 16; off; off >>= 1) ssq += __shfl_xor(ssq, off, 32);
        if (lane == 0) s_red[wave] = ssq;
        __syncthreads();
        if (tid == 0) {
            float t = 0.f;
            for (int w = 0; w < NWAVES; ++w) t += s_red[w];   // fixed order: deterministic
            s_sc[0] = 1.f / sqrtf(t);
        }
        __syncthreads();
        float rn = s_sc[0];
        #pragma unroll
        for (int v = 0; v < 8; ++v) {
            int i = r0 + v + 8 * hh;
            int a = c0 + m;
            s_x[i * STRIDE + a] = xn[v] * rn;   // each element owned by exactly one lane
        }
        __syncthreads();
    }

    // ---- write assignment (out[1 + i*80 + a]) ---------------------------
    #pragma unroll
    for (int v = 0; v < 8; ++v) {
        int i = r0 + v + 8 * hh;
        int a = c0 + m;
        out[1 + i * NN + a] = s_x[i * STRIDE + a];
    }
}

extern "C" void kernel_launch(void* const* d_in, const int* in_sizes, int n_in,
                              void* d_out, int out_size, void* d_ws, size_t ws_size,
                              hipStream_t stream) {
    (void)in_sizes; (void)n_in; (void)out_size; (void)d_ws; (void)ws_size;
    const float* recon = (const float*)d_in[0];   // recon_grf [80*80] f32
    const float* adj   = (const float*)d_in[1];   // original_grf [80*80] f32
    float* out = (float*)d_out;                   // [0]=loss, [1..6400]=assignment

    const size_t shmem = (size_t)(5 * NN * STRIDE + 4 * NN + 32 + 4) * sizeof(float);
    grf_mpm_kernel<<<1, NTHREADS, shmem, stream>>>(recon, adj, out);
}